// RoI_16286515987056
// MI455X (gfx1250) — compile-verified
//
#include <hip/hip_runtime.h>
#include <cmath>

// ---------------------------------------------------------------------------
// ROI-Align (bilinear crop+resize), bandwidth-bound gather kernel for gfx1250.
//  * feature_map: [1, H, W, C] fp32   (H=W=C=256 in the reference setup)
//  * proposals:   [N, 4] fp32 (x1,y1,x2,y2)
//  * out:         [N, S, S, C] fp32   (S=7)
// One block per (roi, output row). The 2 source rows x 2S source columns
// (4*S pixels x C channels = 28 KB for S=7) are staged into LDS via CDNA5
// async global->LDS copies (ASYNCcnt + s_wait_asynccnt), then blended with
// 128-bit LDS reads and streamed out with non-temporal 128-bit stores
// (protects the feature map's L2 residency; output is write-once).
// Specialized instantiation for S=7, C=256 removes runtime-div SALU overhead.
// ---------------------------------------------------------------------------

#if __has_builtin(__builtin_amdgcn_global_load_async_to_lds_b128)
#define USE_ASYNC_LDS 1
#else
#define USE_ASYNC_LDS 0
#endif

typedef int   v4i __attribute__((__vector_size__(4 * sizeof(int))));
typedef float v4f __attribute__((ext_vector_type(4)));
typedef __attribute__((address_space(1))) v4i* g_v4i_ptr;   // global int4*
typedef __attribute__((address_space(3))) v4i* l_v4i_ptr;   // LDS int4*

__device__ __forceinline__ void stage16(const float* gsrc, float* ldst) {
#if USE_ASYNC_LDS
  // async copy of 16 bytes per lane: global -> LDS, tracked with ASYNCcnt
  __builtin_amdgcn_global_load_async_to_lds_b128(
      (g_v4i_ptr)gsrc, (l_v4i_ptr)ldst, /*offset=*/0, /*cpol=*/0);
#else
  *(v4f*)ldst = *(const v4f*)gsrc;
#endif
}

__device__ __forceinline__ void stage_join() {
#if USE_ASYNC_LDS
#if __has_builtin(__builtin_amdgcn_s_wait_asynccnt)
  __builtin_amdgcn_s_wait_asynccnt(0);
#else
  asm volatile("s_wait_asynccnt 0" ::: "memory");
#endif
#endif
}

// TS/TC > 0: compile-time S/C (divisions fold to shifts, loops unroll).
// TS/TC == 0: runtime fallback.
template <int TS, int TC>
__global__ __launch_bounds__(256) void roi_align_gfx1250(
    const float* __restrict__ fm,     // [H, W, C]
    const float* __restrict__ props,  // [N, 4]
    float* __restrict__ out,          // [N, S, S, C]
    int H, int W, int Crt, int Srt)
{
  const int S = (TS > 0) ? TS : Srt;
  const int C = (TC > 0) ? TC : Crt;

  extern __shared__ float smem[];     // 4*S pixels * C floats (28 KB @ S=7)

  const int tid = threadIdx.x;
  const int blk = blockIdx.x;         // blk = roi * S + sy_index
  const int roi = blk / S;
  const int syi = blk - roi * S;

  // Per-ROI box (uniform across block -> scalar loads / SALU).
  const float bx1 = rintf(props[roi * 4 + 0]);   // jnp.round == nearest-even
  const float by1 = rintf(props[roi * 4 + 1]);
  const float bx2 = rintf(props[roi * 4 + 2]);
  const float by2 = rintf(props[roi * 4 + 3]);
  const float x1 = fmaxf(bx1, 0.0f);
  const float y1 = fmaxf(by1, 0.0f);
  const float x2 = fminf(bx2, (float)(W - 1));
  const float y2 = fminf(by2, (float)(H - 1));
  const float h = y2 - y1;
  const float w = x2 - x1;
  const float hmax = fmaxf(h - 1.0f, 0.0f);
  const float wmax = fmaxf(w - 1.0f, 0.0f);

  // y sample for this output row (uniform per block)
  const float ty  = ((float)syi + 0.5f) / (float)S;
  const float sy  = fminf(fmaxf(ty * h - 0.5f, 0.0f), hmax) + y1;
  const float y0f = floorf(sy);
  const int   y0i = (int)y0f;
  const int   y1i = min(y0i + 1, H - 1);
  const float wy  = sy - y0f;
  const float omwy = 1.0f - wy;

  const int vecPerPix = C >> 2;                // 16-byte chunks per pixel

  // ---- Stage 4*S gathered pixels (rows {y0i,y1i} x cols {x0,x1}[s]) to LDS.
  // LDS pixel index p: p = r*(2S) + 2*s + j ; r=row(0/1), s=x-sample, j=lo/hi.
  const int nPix        = 4 * S;
  const int totalChunks = nPix * vecPerPix;

  for (int g = tid; g < totalChunks; g += 256) {
    const int p = g / vecPerPix;
    const int l = g - p * vecPerPix;           // 16B chunk within pixel
    const int r = p / (2 * S);
    const int k = p - r * (2 * S);
    const int s = k >> 1;
    const int j = k & 1;

    const float tx  = ((float)s + 0.5f) / (float)S;
    const float sxp = fminf(fmaxf(tx * w - 0.5f, 0.0f), wmax) + x1;
    const int   x0i = (int)floorf(sxp);
    const int   col = j ? min(x0i + 1, W - 1) : x0i;
    const int   row = r ? y1i : y0i;

    const float* gsrc = fm + ((size_t)row * W + col) * C + (l << 2);
    float*       ldst = smem + (size_t)p * C + (l << 2);
    stage16(gsrc, ldst);
  }
  stage_join();      // drain this wave's ASYNCcnt (data landed in LDS)
  __syncthreads();   // make all waves' staged pixels visible

  // ---- Blend: lanes cover (x-sample, float4 chunk); 128-bit LDS reads,
  // 128-bit non-temporal global stores (output is never re-read).
  float* orow = out + (((size_t)roi * S + syi) * S) * C;  // [S, C] slab
  const int totalVec = S * vecPerPix;

  for (int g = tid; g < totalVec; g += 256) {
    const int s = g / vecPerPix;
    const int l = g - s * vecPerPix;

    const float tx  = ((float)s + 0.5f) / (float)S;
    const float sxp = fminf(fmaxf(tx * w - 0.5f, 0.0f), wmax) + x1;
    const float wx  = sxp - floorf(sxp);
    const float omwx = 1.0f - wx;

    const float* base = smem + (size_t)(2 * s) * C + (l << 2);
    const v4f v00 = *(const v4f*)(base);
    const v4f v01 = *(const v4f*)(base + C);
    const v4f v10 = *(const v4f*)(base + (size_t)(2 * S) * C);
    const v4f v11 = *(const v4f*)(base + (size_t)(2 * S) * C + C);

    const v4f top = v00 * omwx + v01 * wx;
    const v4f bot = v10 * omwx + v11 * wx;
    const v4f res = top * omwy + bot * wy;

    __builtin_nontemporal_store(res, (v4f*)(orow + (size_t)s * C + (l << 2)));
  }
}

extern "C" void kernel_launch(void* const* d_in, const int* in_sizes, int n_in,
                              void* d_out, int out_size, void* d_ws, size_t ws_size,
                              hipStream_t stream) {
  (void)d_ws; (void)ws_size; (void)n_in;

  const float* fm    = (const float*)d_in[0];
  const float* props = (const float*)d_in[1];
  float*       out   = (float*)d_out;

  const int N = in_sizes[1] / 4;          // 1024 proposals
  const int C = 256;                      // channels-last, per reference setup
  const int HW = in_sizes[0] / C;         // H*W
  int W = (int)(sqrt((double)HW) + 0.5);  // square feature map -> 256
  if (W <= 0) W = 256;
  const int H = HW / W;

  // S from out_size = N * S * S * C  (roi_size lives in device memory only)
  int S = 7;
  const long long denom = (long long)N * (long long)C;
  if (denom > 0) {
    const double ss = (double)out_size / (double)denom;
    const int r = (int)(sqrt(ss) + 0.5);
    if (r > 0) S = r;
  }

  const dim3 grid(N * S);
  const dim3 block(256);
  const size_t smemBytes = (size_t)4 * S * C * sizeof(float);  // 28 KB @ S=7

  if (S == 7 && C == 256) {
    roi_align_gfx1250<7, 256><<<grid, block, smemBytes, stream>>>(
        fm, props, out, H, W, C, S);
  } else {
    roi_align_gfx1250<0, 0><<<grid, block, smemBytes, stream>>>(
        fm, props, out, H, W, C, S);
  }
}